// ResonatorBank_22462678958732
// MI455X (gfx1250) — compile-verified
//
#include <hip/hip_runtime.h>
#include <math.h>

// ---------------- problem constants (match reference) ----------------
#define NH      128
#define NN      64
#define NT      32
#define NOSC    224          // NH+NN+NT
#define BATCH   32
#define TSTEPS  1000
#define FSZ     512
#define SRATE   48000.0f
#define DTF     (512.0f / 48000.0f)
#define TWO_PI_F 6.28318530717958647692f
#define INV_2PI  0.15915494309189533577f
#define HALF_PI  1.57079632679489661923f

#define STATE_F (3 * NOSC)                 // 672 floats / frame / batch
#define STATE_B (STATE_F * 4)              // 2688 bytes

typedef __attribute__((ext_vector_type(2))) float v2f;
typedef __attribute__((ext_vector_type(8))) float v8f;
typedef __attribute__((ext_vector_type(4))) int   v4i;
typedef __attribute__((ext_vector_type(8))) int   v8i;

// =====================================================================
// Pass 1: sequential state scan over a chunk of T.  One thread per (b,n),
// spread as (BATCH x 7) blocks of 32 threads for WGP coverage.
//   carry : ws[0 .. BATCH*STATE_B)          persistent scan state
//   states: per-chunk frames, (tl*BATCH + b)*STATE_F + {0,1,2}*NOSC + n
// =====================================================================
__global__ void __launch_bounds__(32)
resobank_state_scan(const float* __restrict__ f0,
                    const float* __restrict__ vel,
                    const float* __restrict__ control,   // (B,T,N,2)
                    const float* __restrict__ gates,     // (B,T,N)
                    const int*   __restrict__ labels,    // (B,T)
                    const float* __restrict__ init_ph,   // (B,N)
                    const float* __restrict__ jitter,    // (T,B,NN)
                    const float* __restrict__ log_inh,   // scalar
                    const float* __restrict__ log_decay, // (N)
                    float*       __restrict__ carry,
                    float*       __restrict__ states,
                    int t0, int tc)
{
    const int b = blockIdx.x;
    const int n = blockIdx.y * 32 + threadIdx.x;
    if (n >= NOSC) return;

    float fr, am, ph;
    if (t0 == 0) {
        // ---- init_state ----
        const float f0b = f0[b];
        if (n < NH) {
            const float i   = (float)(n + 1);
            const float inh = expf(log_inh[0]);
            fr = f0b * i * sqrtf(1.0f + inh * i * i);
        } else if (n < NH + NN) {
            const int j = n - NH;
            fr = expf(logf(200.0f) + (float)j * (logf(10000.0f) - logf(200.0f)) * (1.0f / 63.0f));
        } else {
            const int j = n - (NH + NN);
            fr = f0b * expf(logf(2.0f) + (float)j * (logf(8.0f) - logf(2.0f)) * (1.0f / 31.0f));
        }
        const float amp_t = (n < NH) ? 0.08f / (float)(n + 1)
                                     : ((n < NH + NN) ? 0.005f : 0.03f);
        am = vel[b] * amp_t;
        ph = init_ph[b * NOSC + n];
    } else {
        fr = carry[b * STATE_F + n];
        am = carry[b * STATE_F + NOSC + n];
        ph = carry[b * STATE_F + 2 * NOSC + n];
    }

    const float decayf  = expf(-expf(log_decay[n]) * DTF);
    const bool  isHarm  = (n < NH);
    const bool  isNoise = (n >= NH) && (n < NH + NN);

    for (int tl = 0; tl < tc; ++tl) {
        const int    t    = t0 + tl;
        const size_t cidx = ((size_t)(b * TSTEPS + t) * NOSC + n);
        if (tl + 1 < tc) {   // software prefetch -> global_prefetch_b8
            __builtin_prefetch(&control[(cidx + NOSC) * 2], 0, 0);
            __builtin_prefetch(&gates[cidx + NOSC], 0, 0);
        }
        const float df  = control[cidx * 2 + 0];
        const float da  = control[cidx * 2 + 1];
        const float g   = gates[cidx];
        const int   lab = labels[b * TSTEPS + t];

        // phase mask
        const float sus = (lab == 1 && n >= 12) ? 1.0f : 0.0f;
        const float dcy = (lab == 2 && n >= 6)  ? 1.0f : 0.0f;
        const float rel = (lab == 3)            ? 1.0f : 0.0f;
        const float trn = (lab != 0 && n >= NH + NN) ? 1.0f : 0.0f;
        float mask = 1.0f - sus - dcy - rel - trn;
        mask = fminf(fmaxf(mask, 0.0f), 1.0f);
        const float eff = g * mask;

        if (isHarm)
            fr = fr * (1.0f + 0.01f * tanhf(df) * eff);

        // softplus((amp + da*eff) * decay)
        const float x = (am + da * eff) * decayf;
        am = (x > 20.0f) ? x : log1pf(expf(x));

        // phase advance, wrap to [0, 2pi)
        ph = ph + TWO_PI_F * fr * DTF;
        ph -= floorf(ph * INV_2PI) * TWO_PI_F;
        if (isNoise) {
            ph += jitter[(size_t)(t * BATCH + b) * NN + (n - NH)];
            ph -= floorf(ph * INV_2PI) * TWO_PI_F;
        }

        const size_t base = (size_t)(tl * BATCH + b) * STATE_F;
        states[base + n]            = fr;
        states[base + NOSC + n]     = am;
        states[base + 2 * NOSC + n] = ph;
    }

    carry[b * STATE_F + n]            = fr;
    carry[b * STATE_F + NOSC + n]     = am;
    carry[b * STATE_F + 2 * NOSC + n] = ph;
}

// =====================================================================
// Pass 2: frame synthesis as fp32 WMMA matmul (f = 16q + r):
//   out[q,r] = sum_n sin(ph+16b_n q) * aX_n cos(b_n r)
//            + cos(ph+16b_n q) * aX_n sin(b_n r)
// Frame state staged LDS<-global by the Tensor Data Mover, then phasor
// panels built by 2x2 rotations, then 112 chained V_WMMA_F32_16X16X4_F32
// per wave (wave = channel x M-tile).  K=448 in two halves: LDS ~59 KB.
// =====================================================================
#define KHALF  112            // oscillators per K-phase
#define KP     226            // LDS pitch (224 + 2 pad; 226 % 64 != 0)

__global__ void __launch_bounds__(128)
resobank_synth_wmma(const float* __restrict__ states,
                    const float* __restrict__ pan_params,
                    float*       __restrict__ out,
                    int t0)
{
    __shared__ float SLds[STATE_F];          // TDM-staged frame state, 2688 B
    __shared__ float Amat[32 * KP];          // [q][k]     28928 B
    __shared__ float Bmat[2][16 * KP];       // [ch][r][k] 28928 B

    const int tl  = blockIdx.x;
    const int b   = blockIdx.y;
    const int tid = threadIdx.x;
    const int lane = tid & 31;
    const int wv   = tid >> 5;
    const int ch   = wv >> 1;                // 0 = left, 1 = right
    const int mt   = wv & 1;                 // M tile (q rows 0-15 / 16-31)

    const float* st = states + (size_t)(tl * BATCH + b) * STATE_F;

    // ---- TDM: DMA the 672-dword frame state into LDS ----------------
    {
        const unsigned lds_off = (unsigned)(uintptr_t)(&SLds[0]); // flat addr low 32b == LDS offset
        const unsigned long long ga = (unsigned long long)(uintptr_t)st;
        v4i g0;
        g0.x = 1;                                        // count=1, user descriptor
        g0.y = (int)lds_off;                             // lds_addr
        g0.z = (int)(unsigned)(ga & 0xFFFFFFFFu);        // global_addr[31:0]
        g0.w = (int)(((unsigned)(ga >> 32) & 0x01FFFFFFu) | 0x80000000u); // ga[56:32] | type=2
        v8i g1;
        g1[0] = 0x00020000;                              // data_size = 4 bytes
        g1[1] = (STATE_F & 0xFFFF) << 16;                // tensor_dim0 = 672 (lo16)
        g1[2] = (1 << 16);                               // tensor_dim0 hi=0 | tensor_dim1=1 (lo16)
        g1[3] = (STATE_F << 16);                         // tensor_dim1 hi=0 | tile_dim0 = 672
        g1[4] = 1;                                       // tile_dim1 = 1, tile_dim2 = 0
        g1[5] = STATE_F;                                 // tensor_dim0_stride = 672 (lo32)
        g1[6] = 0;
        g1[7] = 0;
        asm volatile("tensor_load_to_lds %0, %1" :: "s"(g0), "s"(g1) : "memory");
        __builtin_amdgcn_s_wait_tensorcnt(0);
    }

    const int q16  = (lane & 15);            // row within tile / col within tile
    const int koff = (lane >= 16) ? 2 : 0;   // A/B fragment K sub-offset (ISA 7.12.2)
    const int qrow = mt * 16 + q16;

    v8f acc = {};

    for (int p = 0; p < 2; ++p) {
        if (p) __syncthreads();              // phase-0 mma done before overwrite

        // ---- build phasor panels for oscillators [p*112, p*112+112) ----
        if (tid < KHALF) {
            const int   n  = p * KHALF + tid;
            const int   kk = 2 * tid;        // column pair within this K-phase
            const float fr = SLds[n];
            const float am = SLds[NOSC + n];
            const float ph = SLds[2 * NOSC + n];
            const float pan = fminf(fmaxf(pan_params[n], 0.0f), HALF_PI);
            const float aL = am * cosf(pan);
            const float aR = am * sinf(pan);

            const float beta  = TWO_PI_F * fr * (1.0f / SRATE);  // per-sample angle
            const float alpha = 16.0f * beta;                    // per-q angle

            float s, c;   sincosf(ph,    &s,  &c);
            float ds, dc; sincosf(alpha, &ds, &dc);
            #pragma unroll
            for (int q = 0; q < 32; ++q) {
                Amat[q * KP + kk]     = s;   // sin(phi_q)
                Amat[q * KP + kk + 1] = c;   // cos(phi_q)
                const float s2 = s * dc + c * ds;
                c = c * dc - s * ds;
                s = s2;
            }

            float rs, rc; sincosf(beta, &rs, &rc);
            float sr = 0.0f, cr = 1.0f;      // sin/cos(beta*r), r = 0
            #pragma unroll
            for (int r = 0; r < 16; ++r) {
                Bmat[0][r * KP + kk]     = aL * cr;
                Bmat[0][r * KP + kk + 1] = aL * sr;
                Bmat[1][r * KP + kk]     = aR * cr;
                Bmat[1][r * KP + kk + 1] = aR * sr;
                const float s2 = sr * rc + cr * rs;
                cr = cr * rc - sr * rs;
                sr = s2;
            }
        }
        __syncthreads();

        // ---- 56 chained V_WMMA_F32_16X16X4_F32 per phase per wave ----
        const float* Bch = Bmat[ch];
        for (int k0 = 0; k0 < 2 * KHALF; k0 += 4) {
            v2f a, bb;
            a.x  = Amat[qrow * KP + k0 + koff];       // A[m, k0+koff]
            a.y  = Amat[qrow * KP + k0 + koff + 1];   // A[m, k0+koff+1]
            bb.x = Bch[q16 * KP + k0 + koff];         // B[k0+koff,   n]
            bb.y = Bch[q16 * KP + k0 + koff + 1];     // B[k0+koff+1, n]
            acc = __builtin_amdgcn_wmma_f32_16x16x4_f32(
                      false, a, false, bb, (short)0, acc, false, false);
        }
    }

    // ---- scatter 16x16 f32 C/D tile (ISA 7.12.2 layout) ----
    const size_t obase = ((size_t)b * 2 + ch) * ((size_t)TSTEPS * FSZ)
                       + (size_t)(t0 + tl) * FSZ;
    const int rhalf = (lane >= 16) ? 8 : 0;
    #pragma unroll
    for (int j = 0; j < 8; ++j) {
        const int qq = mt * 16 + j + rhalf;          // output row (q)
        out[obase + qq * 16 + q16] = acc[j];         // f = 16q + r
    }
}

// =====================================================================
extern "C" void kernel_launch(void* const* d_in, const int* in_sizes, int n_in,
                              void* d_out, int out_size, void* d_ws, size_t ws_size,
                              hipStream_t stream)
{
    const float* f0        = (const float*)d_in[0];
    const float* vel       = (const float*)d_in[1];
    const float* control   = (const float*)d_in[2];
    const float* gates     = (const float*)d_in[3];
    const int*   labels    = (const int*)  d_in[4];
    const float* init_ph   = (const float*)d_in[5];
    const float* jitter    = (const float*)d_in[6];
    const float* log_inh   = (const float*)d_in[7];
    const float* log_decay = (const float*)d_in[8];
    const float* pan       = (const float*)d_in[9];

    float* outp = (float*)d_out;

    // ws layout: [carry: BATCH*STATE_B][frame states: chunkT*BATCH*STATE_B]
    const size_t carryBytes = (size_t)BATCH * STATE_B;     // 86016 B
    const size_t perT       = (size_t)BATCH * STATE_B;     // 86016 B / timestep
    float* carry  = (float*)d_ws;
    float* states = (float*)((char*)d_ws + carryBytes);

    size_t availT = (ws_size > carryBytes) ? (ws_size - carryBytes) / perT : 0;
    int chunkT = (availT >= (size_t)TSTEPS) ? TSTEPS : (int)availT;
    if (chunkT < 1) chunkT = 1;   // requires ws_size >= 2*86016 B

    for (int tt0 = 0; tt0 < TSTEPS; tt0 += chunkT) {
        const int tc = (TSTEPS - tt0 < chunkT) ? (TSTEPS - tt0) : chunkT;

        resobank_state_scan<<<dim3(BATCH, 7), dim3(32), 0, stream>>>(
            f0, vel, control, gates, labels, init_ph, jitter,
            log_inh, log_decay, carry, states, tt0, tc);

        resobank_synth_wmma<<<dim3(tc, BATCH), dim3(128), 0, stream>>>(
            states, pan, outp, tt0);
    }
}